// OptimizedGNNAnomalyVAE_77841987272810
// MI455X (gfx1250) — compile-verified
//
#include <hip/hip_runtime.h>
#include <math.h>
#include <stdint.h>

#define N_NODES 50000
#define N_EDGES 800000
#define IN_DIM  64
#define HID     96
#define LAT     32
#define BN_EPS  1e-5f
#define NEG_SLOPE 0.2f

typedef float v2f __attribute__((ext_vector_type(2)));
typedef float v8f __attribute__((ext_vector_type(8)));
typedef unsigned int u32x4 __attribute__((ext_vector_type(4)));
typedef int i32x8 __attribute__((ext_vector_type(8)));
typedef int i32x4 __attribute__((ext_vector_type(4)));

// ---------------------------------------------------------------------------
// Generic fp32 WMMA GEMM: C[M,N] = A[M,K] @ B[K,N] (+bias) (+optional min(.,10))
// One 16x16 output tile per wave32, K stepped by 4 via V_WMMA_F32_16X16X4_F32.
// B (K x N, <= 96x96 fp32 = 36 KB) is staged into LDS once per block by the
// Tensor Data Mover (TENSOR_LOAD_TO_LDS, tracked by TENSORcnt), so the hot
// K-loop feeds B fragments from ds_load instead of 16-way-replicated L2 reads.
// Requires M%16==0, N%16==0, K%4==0 (true for all shapes here).
//
// Fragment layouts (CDNA5 ISA 7.12.2):
//   A 16x4 f32 : 2 VGPRs; lanes 0-15 row M=lane, K={0,1}; lanes 16-31 K={2,3}
//   B 4x16 f32 : 2 VGPRs; col = lane&15, K rows split by lane half (mirrors A)
//   C 16x16 f32: 8 VGPRs; VGPR r -> row r (lanes 0-15) / row r+8 (lanes 16-31)
// ---------------------------------------------------------------------------
template <int EPI> // 0: +bias ; 1: +bias then min(v, 10)
__global__ void wmma_gemm_f32(const float* __restrict__ A,
                              const float* __restrict__ B,
                              const float* __restrict__ bias,
                              float* __restrict__ C,
                              int M, int N, int K) {
  __shared__ float ldsB[HID * HID];   // max 96x96 fp32 = 36 KB of the 320 KB WGP LDS

  // ---- TDM stage: one wave issues the DMA of the whole B matrix ----
  if (threadIdx.x < 32) {
    unsigned lds_off = (unsigned)(uintptr_t)(void*)ldsB;   // low 32 bits = LDS byte offset
    unsigned long long ga = (unsigned long long)(uintptr_t)B;
    unsigned uN = (unsigned)N, uK = (unsigned)K;
    u32x4 g0;
    g0[0] = 1u;                                            // count = 1 valid descriptor
    g0[1] = lds_off;                                       // lds_addr
    g0[2] = (unsigned)(ga & 0xFFFFFFFFu);                  // global_addr[31:0]
    g0[3] = (unsigned)((ga >> 32) & 0x1FFFFFFu) | (2u << 30); // addr[56:32] | type=2
    i32x8 g1;
    g1[0] = (int)(2u << 16);                               // data_size = 2 -> 4 bytes
    g1[1] = (int)((uN & 0xFFFFu) << 16);                   // tensor_dim0[15:0]
    g1[2] = (int)(((uN >> 16) & 0xFFFFu) | ((uK & 0xFFFFu) << 16)); // dim0 hi | dim1 lo
    g1[3] = (int)(((uK >> 16) & 0xFFFFu) | ((uN & 0xFFFFu) << 16)); // dim1 hi | tile_dim0=N
    g1[4] = (int)(uK & 0xFFFFu);                           // tile_dim1 = K (tile_dim2 = 0)
    g1[5] = (int)uN;                                       // tensor_dim0_stride lo32 = N
    g1[6] = 0;                                             // stride hi16 | dim1_stride = 0
    g1[7] = 0;
    i32x4 gz4 = {0, 0, 0, 0};                              // 2-D tensor: groups 2/3 unused
    i32x8 gz8 = {0, 0, 0, 0, 0, 0, 0, 0};
    // 6-arg form (clang-23 / therock-10.0 headers): (g0, g1, g2, g3, g4, cpol)
    __builtin_amdgcn_tensor_load_to_lds(g0, g1, gz4, gz4, gz8, 0);
    __builtin_amdgcn_s_wait_tensorcnt(0);                  // TDM done for this wave
  }
  __syncthreads();                                         // publish LDS tile to all waves

  int wid  = (blockIdx.x * blockDim.x + threadIdx.x) >> 5;
  int lane = threadIdx.x & 31;
  int tilesN = N >> 4;
  int tilesM = M >> 4;
  int tm = wid / tilesN;
  int tn = wid - tm * tilesN;
  if (tm >= tilesM) return;

  int half = lane >> 4;          // 0: lanes 0-15, 1: lanes 16-31
  int l15  = lane & 15;
  int arow = (tm << 4) + l15;    // A row held by this lane
  int bcol = (tn << 4) + l15;    // B / C column held by this lane

  const float* arp = A + (long)arow * K;
  v8f acc = {0.f, 0.f, 0.f, 0.f, 0.f, 0.f, 0.f, 0.f};

  for (int k = 0; k < K; k += 4) {
    int kk = k + (half << 1);
    v2f a, b;
    a[0] = arp[kk];                // merges into global_load_b64
    a[1] = arp[kk + 1];
    b[0] = ldsB[kk * N + bcol];    // ds_load_b32 from staged tile
    b[1] = ldsB[(kk + 1) * N + bcol];
    acc = __builtin_amdgcn_wmma_f32_16x16x4_f32(false, a, false, b,
                                                (short)0, acc, false, false);
  }

  float bv = bias ? bias[bcol] : 0.f;
#pragma unroll
  for (int r = 0; r < 8; ++r) {
    int crow = (tm << 4) + r + (half << 3);
    float v = acc[r] + bv;
    if (EPI == 1) v = fminf(v, 10.0f);
    C[(long)crow * N + bcol] = v;
  }
}

// ---------------------------------------------------------------------------
// Per-layer scratch init: m = -inf, s = 0, stats = 0, agg[i] = bias[i % HID]
// (agg pre-loaded with bias so edge scatter accumulates out+bias directly)
// ---------------------------------------------------------------------------
__global__ void layer_init(float* __restrict__ m, float* __restrict__ s,
                           float* __restrict__ agg, const float* __restrict__ bias,
                           float* __restrict__ stats) {
  int tid = blockIdx.x * blockDim.x + threadIdx.x;
  int stride = gridDim.x * blockDim.x;
  for (int i = tid; i < 2 * HID; i += stride) stats[i] = 0.f;
  for (int i = tid; i < N_NODES; i += stride) {
    m[i] = -__builtin_inff();
    s[i] = 0.f;
  }
  for (int i = tid; i < N_NODES * HID; i += stride) agg[i] = bias[i % HID];
}

__device__ __forceinline__ void atomicMaxFloat(float* addr, float val) {
  // sign-split trick: correct total order on IEEE floats, native int atomics
  if (val >= 0.f)
    atomicMax((int*)addr, __float_as_int(val));
  else
    atomicMin((unsigned int*)addr, __float_as_uint(val));
}

// ---------------------------------------------------------------------------
// Pass 1: e = att . leaky_relu(xl[src] + xr[dst]); segment max over dst.
// One wave per edge; lanes stride the 96 features (coalesced row reads, L2 hit)
// ---------------------------------------------------------------------------
__global__ void edge_attn(const float* __restrict__ xl, const float* __restrict__ xr,
                          const float* __restrict__ att,
                          const int* __restrict__ src, const int* __restrict__ dst,
                          float* __restrict__ e, float* __restrict__ m) {
  int wid  = (blockIdx.x * blockDim.x + threadIdx.x) >> 5;
  int lane = threadIdx.x & 31;
  if (wid >= N_EDGES) return;
  int sN = src[wid], dN = dst[wid];
  const float* xls = xl + (long)sN * HID;
  const float* xrd = xr + (long)dN * HID;
  float acc = 0.f;
#pragma unroll
  for (int f = lane; f < HID; f += 32) {
    float v = xls[f] + xrd[f];
    v = v > 0.f ? v : NEG_SLOPE * v;
    acc += v * att[f];
  }
#pragma unroll
  for (int off = 16; off > 0; off >>= 1) acc += __shfl_xor(acc, off, 32);
  if (lane == 0) {
    e[wid] = acc;
    atomicMaxFloat(&m[dN], acc);
  }
}

// Pass 2: p = exp(e - m[dst]); segment sum over dst. Thread per edge.
__global__ void edge_softmax_num(const int* __restrict__ dst,
                                 const float* __restrict__ m,
                                 float* __restrict__ e, float* __restrict__ s) {
  int stride = gridDim.x * blockDim.x;
  for (int i = blockIdx.x * blockDim.x + threadIdx.x; i < N_EDGES; i += stride) {
    int dN = dst[i];
    float p = __expf(e[i] - m[dN]);
    e[i] = p;
    atomicAdd(&s[dN], p);
  }
}

// Pass 3: agg[dst] += (p / (s[dst]+eps)) * xl[src]. Wave per edge, lane-strided.
__global__ void edge_scatter(const float* __restrict__ xl, const float* __restrict__ p,
                             const float* __restrict__ s,
                             const int* __restrict__ src, const int* __restrict__ dst,
                             float* __restrict__ agg) {
  int wid  = (blockIdx.x * blockDim.x + threadIdx.x) >> 5;
  int lane = threadIdx.x & 31;
  if (wid >= N_EDGES) return;
  int sN = src[wid], dN = dst[wid];
  float w = p[wid] / (s[dN] + 1e-16f);
  const float* xls = xl + (long)sN * HID;
  float* aggd = agg + (long)dN * HID;
#pragma unroll
  for (int f = lane; f < HID; f += 32) atomicAdd(&aggd[f], w * xls[f]);
}

// ---------------------------------------------------------------------------
// BatchNorm stats: per-feature sum & sumsq, LDS-privatized then global atomics
// ---------------------------------------------------------------------------
__global__ void bn_stats(const float* __restrict__ h, float* __restrict__ stats) {
  __shared__ float ssum[HID];
  __shared__ float ssq[HID];
  for (int i = threadIdx.x; i < HID; i += blockDim.x) { ssum[i] = 0.f; ssq[i] = 0.f; }
  __syncthreads();
  int stride = gridDim.x * blockDim.x;
  for (int i = blockIdx.x * blockDim.x + threadIdx.x; i < N_NODES * HID; i += stride) {
    float v = h[i];
    int f = i % HID;
    atomicAdd(&ssum[f], v);       // ds_add_f32
    atomicAdd(&ssq[f], v * v);
  }
  __syncthreads();
  for (int i = threadIdx.x; i < HID; i += blockDim.x) {
    atomicAdd(&stats[i], ssum[i]);
    atomicAdd(&stats[HID + i], ssq[i]);
  }
}

// Fused normalize (train-mode, biased var) + affine + exact erf GELU
__global__ void bn_gelu(const float* __restrict__ agg, const float* __restrict__ stats,
                        const float* __restrict__ gamma, const float* __restrict__ beta,
                        float* __restrict__ hout) {
  int stride = gridDim.x * blockDim.x;
  const float invN = 1.0f / (float)N_NODES;
  for (int i = blockIdx.x * blockDim.x + threadIdx.x; i < N_NODES * HID; i += stride) {
    int f = i % HID;
    float mean = stats[f] * invN;
    float var  = stats[HID + f] * invN - mean * mean;
    float rstd = rsqrtf(var + BN_EPS);
    float v = (agg[i] - mean) * rstd * gamma[f] + beta[f];
    hout[i] = 0.5f * v * (1.0f + erff(v * 0.70710678118654752f));
  }
}

__global__ void copy_f32(const float* __restrict__ a, float* __restrict__ b, int n) {
  int stride = gridDim.x * blockDim.x;
  for (int i = blockIdx.x * blockDim.x + threadIdx.x; i < n; i += stride) b[i] = a[i];
}

// ---------------------------------------------------------------------------
extern "C" void kernel_launch(void* const* d_in, const int* in_sizes, int n_in,
                              void* d_out, int out_size, void* d_ws, size_t ws_size,
                              hipStream_t stream) {
  (void)in_sizes; (void)n_in; (void)out_size; (void)ws_size;
  // Input order: x, edge_index, Wl[0..2], Wr[0..2], att[0..2], bias[0..2],
  //              gamma[0..2], beta[0..2], W_mu, b_mu, W_ls, b_ls, W_dec, b_dec
  const float* x   = (const float*)d_in[0];
  const int*   ei  = (const int*)d_in[1];
  const int*   src = ei;
  const int*   dst = ei + N_EDGES;
  const float* Wl[3]    = {(const float*)d_in[2],  (const float*)d_in[3],  (const float*)d_in[4]};
  const float* Wr[3]    = {(const float*)d_in[5],  (const float*)d_in[6],  (const float*)d_in[7]};
  const float* att[3]   = {(const float*)d_in[8],  (const float*)d_in[9],  (const float*)d_in[10]};
  const float* bias[3]  = {(const float*)d_in[11], (const float*)d_in[12], (const float*)d_in[13]};
  const float* gamma[3] = {(const float*)d_in[14], (const float*)d_in[15], (const float*)d_in[16]};
  const float* beta[3]  = {(const float*)d_in[17], (const float*)d_in[18], (const float*)d_in[19]};
  const float* W_mu  = (const float*)d_in[20];
  const float* b_mu  = (const float*)d_in[21];
  const float* W_ls  = (const float*)d_in[22];
  const float* b_ls  = (const float*)d_in[23];
  const float* W_dec = (const float*)d_in[24];
  const float* b_dec = (const float*)d_in[25];

  // Workspace arena (~80.4 MB of floats)
  float* ws    = (float*)d_ws;
  float* h     = ws;                      // 4.8M  activations
  float* xl    = h    + N_NODES * HID;    // 4.8M
  float* xr    = xl   + N_NODES * HID;    // 4.8M
  float* agg   = xr   + N_NODES * HID;    // 4.8M  (init = bias, scatter target)
  float* ebuf  = agg  + N_NODES * HID;    // 0.8M  e, then p (in place)
  float* mbuf  = ebuf + N_EDGES;          // 50K   segment max
  float* sbuf  = mbuf + N_NODES;          // 50K   segment sum
  float* stats = sbuf + N_NODES;          // 192   BN sum / sumsq

  const int TPB = 256;               // 8 wave32s per block
  const int WPB = TPB / 32;
  const int gemmTiles   = (N_NODES / 16) * (HID / 16);
  const int gemmBlocks  = (gemmTiles + WPB - 1) / WPB;
  const int edgeBlocks  = (N_EDGES + WPB - 1) / WPB;   // wave per edge

  for (int l = 0; l < 3; ++l) {
    const float* hin = (l == 0) ? x : h;
    const int K = (l == 0) ? IN_DIM : HID;
    wmma_gemm_f32<0><<<gemmBlocks, TPB, 0, stream>>>(hin, Wl[l], nullptr, xl,
                                                     N_NODES, HID, K);
    wmma_gemm_f32<0><<<gemmBlocks, TPB, 0, stream>>>(hin, Wr[l], nullptr, xr,
                                                     N_NODES, HID, K);
    layer_init<<<1024, TPB, 0, stream>>>(mbuf, sbuf, agg, bias[l], stats);
    edge_attn<<<edgeBlocks, TPB, 0, stream>>>(xl, xr, att[l], src, dst, ebuf, mbuf);
    edge_softmax_num<<<2048, TPB, 0, stream>>>(dst, mbuf, ebuf, sbuf);
    edge_scatter<<<edgeBlocks, TPB, 0, stream>>>(xl, ebuf, sbuf, src, dst, agg);
    bn_stats<<<1024, TPB, 0, stream>>>(agg, stats);
    bn_gelu<<<2048, TPB, 0, stream>>>(agg, stats, gamma[l], beta[l], h);
  }

  // Output layout: x_rec [N,64] | z [N,32] | mu [N,32] | logstd [N,32]
  float* out  = (float*)d_out;
  float* xrec = out;
  float* z    = out  + N_NODES * IN_DIM;
  float* mu   = z    + N_NODES * LAT;
  float* ls   = mu   + N_NODES * LAT;

  const int latTiles = (N_NODES / 16) * (LAT / 16);
  const int latBlocks = (latTiles + WPB - 1) / WPB;
  wmma_gemm_f32<0><<<latBlocks, TPB, 0, stream>>>(h, W_mu, b_mu, mu, N_NODES, LAT, HID);
  wmma_gemm_f32<1><<<latBlocks, TPB, 0, stream>>>(h, W_ls, b_ls, ls, N_NODES, LAT, HID);
  copy_f32<<<2048, TPB, 0, stream>>>(mu, z, N_NODES * LAT);   // z = mu (eval mode)

  const int decTiles = (N_NODES / 16) * (IN_DIM / 16);
  const int decBlocks = (decTiles + WPB - 1) / WPB;
  wmma_gemm_f32<0><<<decBlocks, TPB, 0, stream>>>(mu, W_dec, b_dec, xrec,
                                                  N_NODES, IN_DIM, LAT);
}